// MambaRegressor_23338852286919
// MI455X (gfx1250) — compile-verified
//
#include <hip/hip_runtime.h>
#include <math.h>

typedef float v2f __attribute__((ext_vector_type(2)));
typedef float v8f __attribute__((ext_vector_type(8)));

#define NSEQ  256      // B*C sequences
#define LSEQ  127      // patches
#define DMm   96
#define DIi   192
#define DSs   16
#define DRr   6
#define XPN   38       // DR + 2*DS
#define INPN  384      // 2*DI
#define BATCH 4
#define KDIM  512
#define DINd  64
#define CCH   64
#define PLp   8
#define STRD  4
#define CKc   4

#define LDKMAX 196     // max Kd (192) + 4 padding words (bank-conflict-free)

__device__ __forceinline__ float silu_f(float x) {
    return x / (1.0f + __expf(-x));
}

// ---------------------------------------------------------------------------
// Global -> LDS staging. Prefer the CDNA5 async global->LDS DMA path
// (GLOBAL_LOAD_ASYNC_TO_LDS_B128, ASYNCcnt); fall back to VGPR copy.
// The builtin wants v4i pointers in AS1 (global) / AS3 (LDS).
// ---------------------------------------------------------------------------
#if __has_builtin(__builtin_amdgcn_global_load_async_to_lds_b128)
#define HAVE_ASYNC_LDS 1
typedef int v4i_t __attribute__((vector_size(4 * sizeof(int))));
typedef v4i_t __attribute__((address_space(1)))* g4ptr;
typedef v4i_t __attribute__((address_space(3)))* l4ptr;
#endif

__device__ __forceinline__ void lds_copy16(const float* g, float* l) {
#ifdef HAVE_ASYNC_LDS
    __builtin_amdgcn_global_load_async_to_lds_b128((g4ptr)(g), (l4ptr)(l), 0, 0);
#else
    *(float4*)l = *(const float4*)g;
#endif
}

__device__ __forceinline__ void async_join() {
#ifdef HAVE_ASYNC_LDS
#if __has_builtin(__builtin_amdgcn_s_wait_asynccnt)
    __builtin_amdgcn_s_wait_asynccnt(0);
#else
    asm volatile("s_wait_asynccnt 0" ::: "memory");
#endif
#endif
}

// ---------------------------------------------------------------------------
// WMMA GEMM: Out[M,N] (+)= A[M,Kd] * W[N,Kd]^T (+ bias[n])
// fp32 exact via V_WMMA_F32_16X16X4_F32.
// Block = 128 threads (4 waves) -> 64x16 macro-tile, one 16x16 tile per wave.
// A/W tiles staged in LDS with async b128 copies; padded stride Kd+4 keeps
// the 16-row ds_load_b64 reads bank-conflict-free.
// M must be a multiple of 64, Kd of 4; N may be ragged (masked).
// ---------------------------------------------------------------------------
__global__ __launch_bounds__(128) void gemm_wmma_kernel(
    const float* __restrict__ A, const float* __restrict__ W,
    const float* __restrict__ bias, float* __restrict__ Out,
    int M, int N, int Kd, int accumulate)
{
    __shared__ float As[64 * LDKMAX];
    __shared__ float Ws[16 * LDKMAX];

    const int tid  = threadIdx.x;
    const int wv   = tid >> 5;
    const int lane = tid & 31;
    const int r    = lane & 15;
    const int hi   = lane >> 4;
    const int Mb0  = blockIdx.x * 64;
    const int Nb   = blockIdx.y * 16;
    const int col  = Nb + r;
    const int LDK  = Kd + 4;
    const int KC   = Kd >> 2;          // 16-byte chunks per row

    // ---- stage W tile (16 x Kd), rows clamped for ragged N ----
    for (int i = tid; i < 16 * KC; i += 128) {
        int mm = i / KC, cc = i - mm * KC;
        int wrow = Nb + mm; if (wrow >= N) wrow = N - 1;
        lds_copy16(W + (size_t)wrow * Kd + cc * 4, &Ws[mm * LDK + cc * 4]);
    }
    // ---- stage A tile (64 x Kd) ----
    for (int i = tid; i < 64 * KC; i += 128) {
        int mm = i / KC, cc = i - mm * KC;
        lds_copy16(A + (size_t)(Mb0 + mm) * Kd + cc * 4, &As[mm * LDK + cc * 4]);
    }

    const float msk = (col < N) ? 1.0f : 0.0f;
    v8f acc = {0.f, 0.f, 0.f, 0.f, 0.f, 0.f, 0.f, 0.f};
    if (accumulate) {
#pragma unroll
        for (int j = 0; j < 8; ++j) {
            int row = Mb0 + wv * 16 + j + 8 * hi;
            acc[j] = (col < N) ? Out[(size_t)row * N + col] : 0.0f;
        }
    }

    async_join();
    __syncthreads();

    // ---- 2-stage pipelined WMMA over K ----
    const float* ap = &As[(wv * 16 + r) * LDK + 2 * hi];
    const float* bp = &Ws[r * LDK + 2 * hi];
    v2f a_cur, b_cur;
    a_cur.x = ap[0];        a_cur.y = ap[1];
    b_cur.x = bp[0] * msk;  b_cur.y = bp[1] * msk;
    for (int k0 = 4; k0 < Kd; k0 += 4) {
        v2f a_nxt, b_nxt;
        a_nxt.x = ap[k0];           a_nxt.y = ap[k0 + 1];
        b_nxt.x = bp[k0] * msk;     b_nxt.y = bp[k0 + 1] * msk;
        acc = __builtin_amdgcn_wmma_f32_16x16x4_f32(
            false, a_cur, false, b_cur, (short)0, acc, false, false);
        a_cur = a_nxt; b_cur = b_nxt;
    }
    acc = __builtin_amdgcn_wmma_f32_16x16x4_f32(
        false, a_cur, false, b_cur, (short)0, acc, false, false);

    float bv = 0.0f;
    if (bias != nullptr && col < N) bv = bias[col];
#pragma unroll
    for (int j = 0; j < 8; ++j) {
        int row = Mb0 + wv * 16 + j + 8 * hi;
        if (col < N) Out[(size_t)row * N + col] = acc[j] + bv;
    }
}

// ---------------------------------------------------------------------------
// Patch embed: e[(b*64+c), l, m] = sum_p embed_w[m,p]*h[b, l*4+p, c]
//                                 + embed_b[m] + pos_emb[l,m]
// hproj layout: [(b*512+k)][c]  (GEMM output, row-major)
// ---------------------------------------------------------------------------
__global__ __launch_bounds__(DMm) void embed_kernel(
    const float* __restrict__ hproj, const float* __restrict__ embed_w,
    const float* __restrict__ embed_b, const float* __restrict__ pos_emb,
    float* __restrict__ e)
{
    const int m  = threadIdx.x;            // 0..95
    const int nl = blockIdx.x;             // 0..NSEQ*LSEQ-1
    const int n  = nl / LSEQ;
    const int l  = nl % LSEQ;
    const int b  = n / CCH;
    const int c  = n % CCH;
    float s = 0.0f;
#pragma unroll
    for (int p = 0; p < PLp; ++p) {
        int k = l * STRD + p;
        s += embed_w[m * PLp + p] * hproj[(size_t)(b * KDIM + k) * CCH + c];
    }
    e[(size_t)nl * DMm + m] = s + embed_b[m] + pos_emb[l * DMm + m];
}

// ---------------------------------------------------------------------------
// RMS norm over last dim (96): one wave32 per row, 3 elems/lane.
// ---------------------------------------------------------------------------
__global__ __launch_bounds__(128) void rms_kernel(
    const float* __restrict__ x, const float* __restrict__ w,
    float* __restrict__ out, int rows)
{
    const int wid  = threadIdx.x >> 5;
    const int lane = threadIdx.x & 31;
    const int row  = blockIdx.x * 4 + wid;
    if (row >= rows) return;
    const float* xp = x + (size_t)row * DMm;
    float v0 = xp[lane], v1 = xp[lane + 32], v2 = xp[lane + 64];
    float ss = v0 * v0 + v1 * v1 + v2 * v2;
#pragma unroll
    for (int off = 16; off > 0; off >>= 1) ss += __shfl_xor(ss, off, 32);
    float scale = rsqrtf(ss / (float)DMm + 1e-5f);
    float* op = out + (size_t)row * DMm;
    op[lane]      = v0 * scale * w[lane];
    op[lane + 32] = v1 * scale * w[lane + 32];
    op[lane + 64] = v2 * scale * w[lane + 64];
}

// ---------------------------------------------------------------------------
// Causal depthwise conv (CK=4) over xin = xr[..., :192], then SiLU.
// ---------------------------------------------------------------------------
__global__ __launch_bounds__(256) void conv_silu_kernel(
    const float* __restrict__ xr, const float* __restrict__ cw,
    const float* __restrict__ cb, float* __restrict__ xc)
{
    int t = blockIdx.x * blockDim.x + threadIdx.x;
    if (t >= NSEQ * LSEQ * DIi) return;
    const int d  = t % DIi;
    const int nl = t / DIi;
    const int l  = nl % LSEQ;
    const int n  = nl / LSEQ;
    float s = cb[d];
#pragma unroll
    for (int j = 0; j < CKc; ++j) {
        int lj = l - (CKc - 1) + j;
        if (lj >= 0)
            s += cw[d * CKc + j] * xr[(size_t)(n * LSEQ + lj) * INPN + d];
    }
    xc[(size_t)nl * DIi + d] = silu_f(s);
}

// ---------------------------------------------------------------------------
// Selective scan: one block per sequence (n), one thread per channel d.
// h[16] state lives in registers; sequential over L=127.
// ---------------------------------------------------------------------------
__global__ __launch_bounds__(DIi) void scan_kernel(
    const float* __restrict__ xr,    // [rows][384] (res at offset 192)
    const float* __restrict__ xc,    // [rows][192]
    const float* __restrict__ dbl,   // [rows][38]  (dt|B|C)
    const float* __restrict__ dt_w,  // [192][6]
    const float* __restrict__ dt_b,  // [192]
    const float* __restrict__ A_log, // [192][16]
    const float* __restrict__ Dpar,  // [192]
    float* __restrict__ ybuf)        // [rows][192]
{
    const int d = threadIdx.x;
    const int n = blockIdx.x;

    float dtw[DRr];
#pragma unroll
    for (int r = 0; r < DRr; ++r) dtw[r] = dt_w[d * DRr + r];
    const float dtb = dt_b[d];
    float Av[DSs];
#pragma unroll
    for (int s = 0; s < DSs; ++s) Av[s] = -__expf(A_log[d * DSs + s]);
    const float Dv = Dpar[d];

    float h[DSs];
#pragma unroll
    for (int s = 0; s < DSs; ++s) h[s] = 0.0f;

    for (int l = 0; l < LSEQ; ++l) {
        const size_t row = (size_t)n * LSEQ + l;
        const float* db  = dbl + row * XPN;
        float dtv = dtb;
#pragma unroll
        for (int r = 0; r < DRr; ++r) dtv += db[r] * dtw[r];
        float delta = (dtv > 20.0f) ? dtv : log1pf(__expf(dtv)); // softplus
        float xv  = xc[row * DIi + d];
        float dx  = delta * xv;
        float acc = 0.0f;
#pragma unroll
        for (int s = 0; s < DSs; ++s) {
            float dA = __expf(delta * Av[s]);
            h[s] = dA * h[s] + dx * db[DRr + s];        // B
            acc += h[s] * db[DRr + DSs + s];            // C
        }
        float res = xr[row * INPN + DIi + d];
        float y = acc + xv * Dv;
        y *= silu_f(res);
        ybuf[row * DIi + d] = y;
    }
}

// ---------------------------------------------------------------------------
// fc: y[n] = dot(u[n, 0:12192], fc_w) + fc_b   (block reduction)
// ---------------------------------------------------------------------------
__global__ __launch_bounds__(256) void fc_kernel(
    const float* __restrict__ u, const float* __restrict__ fc_w,
    const float* __restrict__ fc_b, float* __restrict__ yv)
{
    __shared__ float red[256];
    const int n = blockIdx.x;
    const float* up = u + (size_t)n * (LSEQ * DMm);
    float s = 0.0f;
    for (int i = threadIdx.x; i < LSEQ * DMm; i += 256) s += up[i] * fc_w[i];
    red[threadIdx.x] = s;
    __syncthreads();
    for (int off = 128; off > 0; off >>= 1) {
        if (threadIdx.x < off) red[threadIdx.x] += red[threadIdx.x + off];
        __syncthreads();
    }
    if (threadIdx.x == 0) yv[n] = red[0] + fc_b[0];
}

// ---------------------------------------------------------------------------
// head: out[b,o] = sum_c yv[b*64+c]*head_w[o,c] + head_b[o]   (4x2 output)
// ---------------------------------------------------------------------------
__global__ void head_kernel(const float* __restrict__ yv,
                            const float* __restrict__ hw,
                            const float* __restrict__ hb,
                            float* __restrict__ out)
{
    int t = threadIdx.x;
    if (t < 8) {
        int b = t >> 1, o = t & 1;
        float s = hb[o];
        for (int c = 0; c < CCH; ++c) s += yv[b * CCH + c] * hw[o * CCH + c];
        out[b * 2 + o] = s;
    }
}

// ---------------------------------------------------------------------------
extern "C" void kernel_launch(void* const* d_in, const int* in_sizes, int n_in,
                              void* d_out, int out_size, void* d_ws, size_t ws_size,
                              hipStream_t stream)
{
    const float* x         = (const float*)d_in[0];
    const float* proj_w    = (const float*)d_in[1];
    const float* proj_b    = (const float*)d_in[2];
    const float* embed_w   = (const float*)d_in[3];
    const float* embed_b   = (const float*)d_in[4];
    const float* pos_emb   = (const float*)d_in[5];
    const float* norm_w    = (const float*)d_in[6];
    const float* in_proj_w = (const float*)d_in[7];
    const float* conv_w    = (const float*)d_in[8];
    const float* conv_b    = (const float*)d_in[9];
    const float* x_proj_w  = (const float*)d_in[10];
    const float* dt_proj_w = (const float*)d_in[11];
    const float* dt_proj_b = (const float*)d_in[12];
    const float* A_log     = (const float*)d_in[13];
    const float* Dpar      = (const float*)d_in[14];
    const float* out_proj_w= (const float*)d_in[15];
    const float* norm_f_w  = (const float*)d_in[16];
    const float* fc_w      = (const float*)d_in[17];
    const float* fc_b      = (const float*)d_in[18];
    const float* head_w    = (const float*)d_in[19];
    const float* head_b    = (const float*)d_in[20];
    float* out = (float*)d_out;

    const int rows = NSEQ * LSEQ;  // 32512 = 64 * 508

    float* ws = (float*)d_ws;
    size_t off = 0;
    float* e     = ws + off; off += (size_t)rows * DMm;        // 3.12M
    float* u     = ws + off; off += (size_t)rows * DMm;        // 3.12M
    float* hproj = ws + off; off += (size_t)BATCH * KDIM * CCH;// 0.13M
    float* xr    = ws + off; off += (size_t)rows * INPN;       // 12.5M
    float* xc    = ws + off; off += (size_t)rows * DIi;        // 6.2M
    float* dblb  = ws + off; off += (size_t)rows * XPN;        // 1.2M
    float* ybuf  = ws + off; off += (size_t)rows * DIi;        // 6.2M
    float* yv    = ws + off; off += NSEQ;

    // 1) proj: h[b*512+k][c] = x[b,k,:].proj_w[c,:] + proj_b[c]
    gemm_wmma_kernel<<<dim3((BATCH * KDIM) / 64, CCH / 16), 128, 0, stream>>>(
        x, proj_w, proj_b, hproj, BATCH * KDIM, CCH, DINd, 0);

    // 2) patch embed + pos_emb -> e
    embed_kernel<<<rows, DMm, 0, stream>>>(hproj, embed_w, embed_b, pos_emb, e);

    // 3) Mamba layers
    for (int i = 0; i < 3; ++i) {
        rms_kernel<<<(rows + 3) / 4, 128, 0, stream>>>(e, norm_w + i * DMm, u, rows);
        gemm_wmma_kernel<<<dim3(rows / 64, INPN / 16), 128, 0, stream>>>(
            u, in_proj_w + (size_t)i * INPN * DMm, nullptr, xr, rows, INPN, DMm, 0);
        int tot = rows * DIi;
        conv_silu_kernel<<<(tot + 255) / 256, 256, 0, stream>>>(
            xr, conv_w + i * DIi * CKc, conv_b + i * DIi, xc);
        gemm_wmma_kernel<<<dim3(rows / 64, 3), 128, 0, stream>>>(   // N=38 -> 3 tiles
            xc, x_proj_w + (size_t)i * XPN * DIi, nullptr, dblb, rows, XPN, DIi, 0);
        scan_kernel<<<NSEQ, DIi, 0, stream>>>(
            xr, xc, dblb, dt_proj_w + i * DIi * DRr, dt_proj_b + i * DIi,
            A_log + i * DIi * DSs, Dpar + i * DIi, ybuf);
        gemm_wmma_kernel<<<dim3(rows / 64, DMm / 16), 128, 0, stream>>>( // e += y@W^T
            ybuf, out_proj_w + (size_t)i * DMm * DIi, nullptr, e, rows, DMm, DIi, 1);
    }

    // 4) final norm + fc + head
    rms_kernel<<<(rows + 3) / 4, 128, 0, stream>>>(e, norm_f_w, u, rows);
    fc_kernel<<<NSEQ, 256, 0, stream>>>(u, fc_w, fc_b, yv);
    head_kernel<<<1, 32, 0, stream>>>(yv, head_w, head_b, out);
}